// RoIPolling_34883724378589
// MI455X (gfx1250) — compile-verified
//
#include <hip/hip_runtime.h>
#include <hip/hip_bf16.h>
#include <stdint.h>

// RoI max pooling for MI455X (gfx1250).
// CDNA5 path: async global->LDS staging (ASYNCcnt) of each ROI rectangle,
// then LDS-resident 7x7 max pooling. WMMA is inapplicable (max-reduction).

#define OUT    7
#define CPB    4      // channels per block
#define MAXS   56     // max ROI side
#define ROWP   57     // padded LDS row stride in floats (bank spread)
#define CH     256
#define FH     128
#define FW     128

typedef __attribute__((address_space(1))) int as1_int;
typedef __attribute__((address_space(3))) int as3_int;

__device__ __forceinline__ void async_g2l_b32(const float* gsrc, float* ldst) {
#if defined(__HIP_DEVICE_COMPILE__) && __has_builtin(__builtin_amdgcn_global_load_async_to_lds_b32)
    __builtin_amdgcn_global_load_async_to_lds_b32(
        (as1_int*)gsrc, (as3_int*)ldst, /*offset=*/0, /*cpol=*/0);
#else
    asm volatile("global_load_async_to_lds_b32 %0, %1, off"
                 :: "v"((uint32_t)(uintptr_t)ldst), "v"(gsrc)
                 : "memory");
#endif
}

__device__ __forceinline__ void wait_asynccnt0() {
#if defined(__HIP_DEVICE_COMPILE__) && __has_builtin(__builtin_amdgcn_s_wait_asynccnt)
    __builtin_amdgcn_s_wait_asynccnt(0);
#else
    asm volatile("s_wait_asynccnt 0x0" ::: "memory");
#endif
}

__global__ __launch_bounds__(256)
void RoIPool_gfx1250_kernel(const float* __restrict__ feats,
                            const int* __restrict__ rois,
                            float* __restrict__ out) {
    __shared__ float tile[CPB * MAXS * ROWP];   // ~51 KB

    const int n   = blockIdx.x;       // ROI index
    const int cb  = blockIdx.y * CPB; // base channel of this block
    const int tid = threadIdx.x;

    // ROI box (uniform across block). Boxes are always fully in-bounds:
    // x <= FW-56, w <= 56  ->  x+w <= FW (same for y/h), so no clipping needed.
    const int x = rois[n * 4 + 0];
    const int y = rois[n * 4 + 1];
    const int w = rois[n * 4 + 2];
    const int h = rois[n * 4 + 3];

    // ---- Stage the CPB x h x w rectangle into LDS with async copies ----
    // Division-free mapping: wave per row (8 waves round-robin over rows),
    // lanes stride columns by 32; each (r,col) address feeds all 4 channels.
    const int wid  = tid >> 5;        // wave id 0..7
    const int lane = tid & 31;
    const float* fbase = feats + (size_t)cb * (FH * FW) + (size_t)y * FW + x;

    for (int r = wid; r < h; r += 8) {
        const float* grow = fbase + r * FW;
        float*       lrow = &tile[r * ROWP];
        for (int col = lane; col < w; col += 32) {
            const float* g = grow + col;
            float*       l = lrow + col;
#pragma unroll
            for (int cl = 0; cl < CPB; ++cl)
                async_g2l_b32(g + cl * (FH * FW), l + cl * (MAXS * ROWP));
        }
    }
    wait_asynccnt0();   // this wave's async LDS writes have landed
    __syncthreads();    // all waves' writes visible block-wide

    // ---- 7x7 max pool per channel (196 active threads) ----
    if (tid < CPB * OUT * OUT) {
        const int cl  = tid / (OUT * OUT);
        const int bin = tid - cl * (OUT * OUT);
        const int oy  = bin / OUT;
        const int ox  = bin - oy * OUT;

        // Exact reference bin bounds (relative to ROI origin).
        const int rsy = (oy * h) / OUT;
        const int rey = ((oy + 1) * h + OUT - 1) / OUT;   // <= h, size <= 8
        const int rsx = (ox * w) / OUT;
        const int rex = ((ox + 1) * w + OUT - 1) / OUT;   // <= w, size <= 8

        float m = -__builtin_huge_valf();                 // bins never empty
        const float* base = &tile[cl * (MAXS * ROWP)];
        for (int r = rsy; r < rey; ++r) {
            const float* row = base + r * ROWP;
            for (int cc = rsx; cc < rex; ++cc)
                m = fmaxf(m, row[cc]);
        }
        // out[n][c][oy][ox]
        out[(((size_t)n * CH + (cb + cl)) * OUT + oy) * OUT + ox] = m;
    }
}

extern "C" void kernel_launch(void* const* d_in, const int* in_sizes, int n_in,
                              void* d_out, int out_size, void* d_ws, size_t ws_size,
                              hipStream_t stream) {
    (void)n_in; (void)out_size; (void)d_ws; (void)ws_size;
    const float* feats = (const float*)d_in[0];
    const int*   rois  = (const int*)d_in[1];
    float*       out   = (float*)d_out;

    const int N = in_sizes[1] / 4;   // 256 ROIs

    dim3 grid((unsigned)N, (unsigned)(CH / CPB));
    RoIPool_gfx1250_kernel<<<grid, 256, 0, stream>>>(feats, rois, out);
}